// PiecewiseQuadratic_23252952940928
// MI455X (gfx1250) — compile-verified
//
#include <hip/hip_runtime.h>

#define NB 32
#define EPSMIN 1e-10f
#define EPSMAX (1.0f - 1e-10f)

typedef __attribute__((ext_vector_type(2))) float v2f;
typedef __attribute__((ext_vector_type(8))) float v8f;

__device__ __forceinline__ float logaddexpf(float a, float b) {
    float mx = fmaxf(a, b);
    float mn = fminf(a, b);
    return mx + log1pf(expf(mn - mx));
}

__launch_bounds__(256)
__global__ void pq_flow_kernel(const float* __restrict__ x_in,
                               const float* __restrict__ width,
                               const float* __restrict__ v_node,
                               float* __restrict__ x_out,
                               float* __restrict__ log_det,
                               int B)
{
    // Per-dimension tables, stride 33 (odd) => 32 lanes with distinct d hit 32 banks.
    __shared__ __align__(16) float sXN[32 * 33];  // x_node, k = 0..32
    __shared__ __align__(16) float sW [32 * 33];  // bin width w
    __shared__ __align__(16) float sIW[32 * 33];  // 1/w
    __shared__ __align__(16) float sV0[32 * 33];  // v0 = exp(v[k]-ynorm)
    __shared__ __align__(16) float sHD[32 * 33];  // 0.5*(v1-v0)
    __shared__ __align__(16) float sYN[32 * 33];  // y_node[:-1]
    __shared__ __align__(16) float sLG[8 * 16 * 32]; // per-wave log-grad staging

    const int tid  = threadIdx.x;
    const int lane = tid & 31;
    const int wid  = tid >> 5;

    // ---------- phase 1: derive per-dimension params (one thread per d) ----------
    if (tid < 32) {
        const int d = tid;
        const float* wrow = width  + d * NB;
        const float* vrow = v_node + d * (NB + 1);
        // softmax over width row (BETA = 1)
        float m = -3.4e38f;
        for (int k = 0; k < NB; ++k) m = fmaxf(m, wrow[k]);
        float s = 0.f;
        for (int k = 0; k < NB; ++k) {
            float e = expf(wrow[k] - m);
            sW[d * 33 + k] = e;
            s += e;
        }
        float inv = 1.f / s;
        float s2 = 0.f;
        for (int k = 0; k < NB; ++k) {  // matches reference's renormalize pass
            float w2 = sW[d * 33 + k] * inv;
            sW[d * 33 + k] = w2;
            s2 += w2;
        }
        float inv2 = 1.f / s2;
        // cumsum -> x_node
        float acc = 0.f;
        sXN[d * 33 + 0] = 0.f;
        for (int k = 0; k < NB; ++k) {
            float w2 = sW[d * 33 + k] * inv2;
            sW [d * 33 + k] = w2;
            sIW[d * 33 + k] = 1.f / w2;
            acc += w2;
            sXN[d * 33 + k + 1] = acc;
        }
        // cumlogsumexp of tmp -> y_log; y_log[0] = -inf (use -1e30)
        float r = 0.f;
        sYN[d * 33 + 0] = -1e30f;
        for (int k = 0; k < NB; ++k) {
            float t = logaddexpf(vrow[k], vrow[k + 1]) + logf(0.5f * sW[d * 33 + k]);
            r = (k == 0) ? t : logaddexpf(r, t);
            sYN[d * 33 + k + 1] = r;
        }
        const float ynorm = r;  // y_log[32]
        for (int k = 0; k < NB; ++k) {
            sYN[d * 33 + k] = expf(sYN[d * 33 + k] - ynorm);
            float v0 = expf(vrow[k]     - ynorm);
            float v1 = expf(vrow[k + 1] - ynorm);
            sV0[d * 33 + k] = v0;
            sHD[d * 33 + k] = 0.5f * (v1 - v0);
        }
    }
    __syncthreads();

    // ---------- phase 2: streaming transform, lane = dimension ----------
    const int rowbase = blockIdx.x * 128 + wid * 16;
    const int d = lane;
    #pragma unroll 4
    for (int r = 0; r < 16; ++r) {
        const int row = rowbase + r;
        // speculative stream prefetch (gfx1250 global_prefetch_b8)
        __builtin_prefetch(x_in + (size_t)(row + 8) * NB + d, 0, 0);
        float lg = 0.f;
        if (row < B) {
            float x = x_in[(size_t)row * NB + d];
            x = fminf(fmaxf(x, EPSMIN), EPSMAX);
            // binary search: largest j in [0,31] with x_node[j] < x
            int j = 0;
            if (sXN[d * 33 + j + 16] < x) j += 16;
            if (sXN[d * 33 + j + 8]  < x) j += 8;
            if (sXN[d * 33 + j + 4]  < x) j += 4;
            if (sXN[d * 33 + j + 2]  < x) j += 2;
            if (sXN[d * 33 + j + 1]  < x) j += 1;
            float w  = sW [d * 33 + j];
            float xl = (x - sXN[d * 33 + j]) * sIW[d * 33 + j];
            float v0 = sV0[d * 33 + j];
            float hd = sHD[d * 33 + j];
            float xo = w * (v0 * xl + hd * xl * xl) + sYN[d * 33 + j];
            // reference sums EPS_MIN over the 31 non-selected bins before the max
            float grad = v0 + 2.f * hd * xl + 31.f * EPSMIN;
            grad = fmaxf(grad, EPSMIN);
            lg = logf(grad);
            x_out[(size_t)row * NB + d] = xo;
        }
        sLG[wid * 512 + r * 32 + lane] = lg;  // lg[row r][dim lane]
    }
    __syncthreads();

    // ---------- phase 3: logdet = row-sum over D=32 via 8x V_WMMA_F32_16X16X4_F32 ----------
    // A (16x32 over 8 chunks of K=4): lane L supplies A[M=L%16][K = 4c + 2*(L/16) + {0,1}]
    const int mrow = lane & 15;
    const int hv   = lane >> 4;
    v8f cacc = {};
    v2f ones = {1.f, 1.f};
    #pragma unroll
    for (int c = 0; c < 8; ++c) {
        const int d0 = 4 * c + 2 * hv;
        v2f a = *(const v2f*)&sLG[wid * 512 + mrow * 32 + d0];
        cacc = __builtin_amdgcn_wmma_f32_16x16x4_f32(false, a, false, ones,
                                                     (short)0, cacc, false, false);
    }
    // D layout: lane L, VGPR t holds rowsum[M = t + 8*(L/16)], identical across all
    // 16 N-columns. Convergent epilogue: each lane selects cacc[lane&7], one wave32
    // shuffle pulls rowsums 8..15 down from the upper half-lanes, then lanes 0..15
    // issue a single coalesced 64B store.
    const int t = lane & 7;
    float sel = cacc[0];
    sel = (t == 1) ? cacc[1] : sel;
    sel = (t == 2) ? cacc[2] : sel;
    sel = (t == 3) ? cacc[3] : sel;
    sel = (t == 4) ? cacc[4] : sel;
    sel = (t == 5) ? cacc[5] : sel;
    sel = (t == 6) ? cacc[6] : sel;
    sel = (t == 7) ? cacc[7] : sel;
    const int src = (lane & 8) ? (lane + 16) : lane;  // rowsums 8..15 live in lanes 16..31
    const float rsum = __shfl(sel, src, 32);
    const int lrow = rowbase + lane;
    if (lane < 16 && lrow < B) log_det[lrow] = rsum;
}

extern "C" void kernel_launch(void* const* d_in, const int* in_sizes, int n_in,
                              void* d_out, int out_size, void* d_ws, size_t ws_size,
                              hipStream_t stream) {
    const float* x_in  = (const float*)d_in[0];
    const float* width = (const float*)d_in[1];
    const float* vnode = (const float*)d_in[2];
    const int B = in_sizes[0] / NB;          // 65536
    float* x_out   = (float*)d_out;          // B*32 floats
    float* log_det = x_out + (size_t)B * NB; // then B floats
    const int grid = (B + 127) / 128;        // 128 rows per 256-thread (8-wave) block
    pq_flow_kernel<<<grid, 256, 0, stream>>>(x_in, width, vnode, x_out, log_det, B);
}